// Channel2D3_38817914421485
// MI455X (gfx1250) — compile-verified
//
#include <hip/hip_runtime.h>

typedef __attribute__((ext_vector_type(2))) float v2f;
typedef __attribute__((ext_vector_type(8))) float v8f;

namespace {
constexpr int N_SRC = 24;
constexpr int N_OUT = 49;
constexpr int BB    = 64;
constexpr int CC_   = 2;
constexpr int SLABS = BB * CC_;          // 128 independent [24 x L] -> [49 x L] GEMMs
constexpr int L     = 32768;
constexpr int WAVES_PER_BLOCK = 8;       // 256 threads (8 wave32)
constexpr int CHUNKS_PER_WAVE = 8;       // each wave covers 8 * 16 = 128 columns of l
constexpr int L_PER_BLOCK     = WAVES_PER_BLOCK * CHUNKS_PER_WAVE * 16; // 1024
constexpr int BLOCKS_PER_SLAB = L / L_PER_BLOCK;                        // 32
}

// Fixed adjacency (1-based source indices, -1 = none), from the reference.
__device__ __constant__ short TRI[N_OUT][4] = {
    {1,4,-1,-1},  {1,-1,-1,-1}, {1,2,5,-1},   {2,-1,-1,-1}, {2,3,6,-1},
    {3,-1,-1,-1}, {3,7,-1,-1},  {4,-1,-1,-1}, {1,4,5,8},    {5,-1,-1,-1},
    {2,5,6,9},    {6,-1,-1,-1}, {3,6,7,10},   {7,-1,-1,-1}, {4,8,11,-1},
    {8,-1,-1,-1}, {5,8,9,12},   {9,-1,-1,-1}, {6,9,10,13},  {10,-1,-1,-1},
    {7,10,14,-1}, {11,-1,-1,-1},{8,11,12,15}, {12,-1,-1,-1},{9,12,13,16},
    {13,-1,-1,-1},{10,13,14,17},{14,-1,-1,-1},{11,15,18,-1},{15,-1,-1,-1},
    {12,15,16,19},{16,-1,-1,-1},{13,16,17,20},{17,-1,-1,-1},{14,17,21,-1},
    {18,-1,-1,-1},{15,18,19,22},{19,-1,-1,-1},{16,19,20,23},{20,-1,-1,-1},
    {17,20,21,24},{21,-1,-1,-1},{18,22,-1,-1},{22,-1,-1,-1},{19,22,23,-1},
    {23,-1,-1,-1},{20,23,24,-1},{24,-1,-1,-1},{21,24,-1,-1}
};

// out[49, L] = W[49, 24] * x[24, L] per slab, using V_WMMA_F32_16X16X4_F32.
// One wave computes a 64(M, padded)x16(N=l) tile, accumulating K=24 in 6 steps of 4.
__global__ __launch_bounds__(WAVES_PER_BLOCK * 32)
void fan_avg_wmma(const float* __restrict__ x, float* __restrict__ out) {
  const int slab = blockIdx.x / BLOCKS_PER_SLAB;
  const int lblk = blockIdx.x % BLOCKS_PER_SLAB;
  const int lane = threadIdx.x & 31;
  const int wv   = threadIdx.x >> 5;
  const int colv = lane & 15;   // N (B/C/D) or M (A) index within 16-wide tile
  const int hi   = lane >> 4;   // lane half: selects K pair for A/B, M+8 for C/D

  const float* __restrict__ xs = x   + (size_t)slab * N_SRC * L;
  float* __restrict__ os       = out + (size_t)slab * N_OUT * L;

  // ---- Build A (W padded to 64x24) fragments in registers, once per wave.
  // 16x4 f32 A layout: VGPR0 holds K=0 (lanes 0-15) / K=2 (lanes 16-31),
  //                    VGPR1 holds K=1 / K=3.
  float A0[4][6], A1[4][6];
#pragma unroll
  for (int mt = 0; mt < 4; ++mt) {
    const int m = mt * 16 + colv;
    int t0 = -2, t1 = -2, t2 = -2, t3 = -2;
    float inv = 0.0f;
    if (m < N_OUT) {
      t0 = (int)TRI[m][0] - 1;
      t1 = (int)TRI[m][1] - 1;
      t2 = (int)TRI[m][2] - 1;
      t3 = (int)TRI[m][3] - 1;
      const int deg = (t0 >= 0) + (t1 >= 0) + (t2 >= 0) + (t3 >= 0);
      inv = 1.0f / (float)deg;
    }
#pragma unroll
    for (int q = 0; q < 6; ++q) {
      const int k0 = q * 4 + hi * 2;
      const int k1 = k0 + 1;
      A0[mt][q] = ((t0 == k0) || (t1 == k0) || (t2 == k0) || (t3 == k0)) ? inv : 0.0f;
      A1[mt][q] = ((t0 == k1) || (t1 == k1) || (t2 == k1) || (t3 == k1)) ? inv : 0.0f;
    }
  }

  // Per-lane base for B loads: B 4x16 layout mirrors A (VGPR0 = K 0|2, VGPR1 = K 1|3),
  // lanes 0-15 / 16-31 each read 64 contiguous bytes of one source row.
  const float* __restrict__ xcol = xs + (size_t)(hi * 2) * L + colv;
  const int lbase = lblk * L_PER_BLOCK + wv * (CHUNKS_PER_WAVE * 16);

  for (int cc = 0; cc < CHUNKS_PER_WAVE; ++cc) {
    const int l0 = lbase + cc * 16;
    v8f acc[4] = {};   // zero C for the 4 M-tiles
#pragma unroll
    for (int q = 0; q < 6; ++q) {
      v2f Bf;
      Bf.x = xcol[(size_t)(q * 4 + 0) * L + l0];  // K = q*4 + 0|2
      Bf.y = xcol[(size_t)(q * 4 + 1) * L + l0];  // K = q*4 + 1|3
#pragma unroll
      for (int mt = 0; mt < 4; ++mt) {
        v2f Af;
        Af.x = A0[mt][q];
        Af.y = A1[mt][q];
        acc[mt] = __builtin_amdgcn_wmma_f32_16x16x4_f32(
            /*neg_a=*/false, Af, /*neg_b=*/false, Bf,
            /*c_mod=*/(short)0, acc[mt], /*reuse_a=*/false, /*reuse_b=*/false);
      }
    }
    // D layout: lanes 0-15: VGPR j -> M=j; lanes 16-31: VGPR j -> M=j+8.
#pragma unroll
    for (int mt = 0; mt < 4; ++mt) {
#pragma unroll
      for (int j = 0; j < 8; ++j) {
        const int m = mt * 16 + hi * 8 + j;
        if (m < N_OUT) os[(size_t)m * L + l0 + colv] = acc[mt][j];
      }
    }
  }
}

extern "C" void kernel_launch(void* const* d_in, const int* in_sizes, int n_in,
                              void* d_out, int out_size, void* d_ws, size_t ws_size,
                              hipStream_t stream) {
  (void)in_sizes; (void)n_in; (void)out_size; (void)d_ws; (void)ws_size;
  const float* x = (const float*)d_in[0];   // [64, 2, 24, 32768] fp32
  float* out     = (float*)d_out;           // [64, 2, 49, 32768] fp32
  dim3 grid(SLABS * BLOCKS_PER_SLAB);       // 4096 blocks
  dim3 block(WAVES_PER_BLOCK * 32);         // 256 threads = 8 wave32
  hipLaunchKernelGGL(fan_avg_wmma, grid, block, 0, stream, x, out);
}